// MatcherSimple_58231166599710
// MI455X (gfx1250) — compile-verified
//
#include <hip/hip_runtime.h>
#include <stdint.h>

// Problem shape (fixed by the reference): B=8, P=4096, G=96.
#define BN 8
#define PN 4096
#define GN 96
#define BIGF 1e9f

typedef __attribute__((ext_vector_type(2))) float v2f;
typedef __attribute__((ext_vector_type(8))) float v8f;

#if defined(__gfx1250__)
typedef __attribute__((address_space(1))) int g1_int_t;  // "__device__ int"
typedef __attribute__((address_space(3))) int l3_int_t;  // "__shared__ int"
#endif

// ---------------- CDNA5 async global->LDS copy (ASYNCcnt-tracked) ----------
__device__ __forceinline__ void async_copy_b32(const float* gsrc, float* ldst) {
#if defined(__gfx1250__)
#if __has_builtin(__builtin_amdgcn_global_load_async_to_lds_b32)
  __builtin_amdgcn_global_load_async_to_lds_b32(
      (g1_int_t*)(void*)gsrc, (l3_int_t*)(void*)ldst, 0, 0);
#else
  asm volatile("global_load_async_to_lds_b32 %0, %1, off"
               :
               : "v"((l3_int_t*)(void*)ldst), "v"(gsrc)
               : "memory");
#endif
#else
  *ldst = *gsrc;  // host-pass / non-gfx1250 stub (never executed on device)
#endif
}

__device__ __forceinline__ void wait_async_zero() {
#if defined(__gfx1250__)
#if __has_builtin(__builtin_amdgcn_s_wait_asynccnt)
  __builtin_amdgcn_s_wait_asynccnt(0);
#else
  asm volatile("s_wait_asynccnt 0" ::: "memory");
#endif
#endif
}

// ---------------- CDNA5 f32 WMMA (16x16x4) ---------------------------------
__device__ __forceinline__ v8f wmma_f32_16x16x4(v2f a, v2f b, v8f c) {
#if defined(__gfx1250__)
  // 8 args: (neg_a, A, neg_b, B, c_mod, C, reuse_a, reuse_b)
  return __builtin_amdgcn_wmma_f32_16x16x4_f32(false, a, false, b, (short)0, c,
                                               false, false);
#else
  return c;  // host-pass stub
#endif
}

// ===========================================================================
// Kernel 1: build costT[b][g][p] = center_dist[b][p][g] - 2*gious[b][p][g]
// Stage a 128(p) x 16(g) tile of both inputs into LDS with async copies,
// then each wave emits one 16(g) x 16(p) output tile with 4 chained
// V_WMMA_F32_16X16X4_F32: D = sum_t A_t x B_t + C, where
//   A_t[g][k] = center[p0+4t+k][g]   (A-layout: lane = g, VGPRs/half = k)
//   B_t[k][p] = (p == 4t+k)          (identity selector)
//   C  [g][p] = -2 * giou[p0+p][g]   (D-layout)
// so D[g][p] = center[p0+p][g] - 2*giou[p0+p][g], the transposed cost tile.
// ===========================================================================
__global__ __launch_bounds__(256) void build_cost_kernel(
    const float* __restrict__ cd, const float* __restrict__ gi,
    float* __restrict__ costT) {
  __shared__ float lds_cd[128 * 16];  // [p_local][g_local]
  __shared__ float lds_gi[128 * 16];

  const int tid = threadIdx.x;
  const int p0 = blockIdx.x * 128;  // gridDim.x = PN/128 = 32
  const int g0 = blockIdx.y * 16;   // gridDim.y = GN/16  = 6
  const int b = blockIdx.z;         // gridDim.z = BN

  // Async-stage the tile: 2048 elems per array, 8 per thread, 64B-contiguous
  // runs of 16 floats per proposal row.
  const size_t ibase = ((size_t)b * PN + (size_t)p0) * GN + (size_t)g0;
#pragma unroll
  for (int k = 0; k < 8; ++k) {
    const int e = tid + 256 * k;
    const int pl = e >> 4;
    const int gl = e & 15;
    const size_t off = ibase + (size_t)pl * GN + (size_t)gl;
    async_copy_b32(cd + off, &lds_cd[e]);
    async_copy_b32(gi + off, &lds_gi[e]);
  }
  wait_async_zero();
  __syncthreads();

  const int w = tid >> 5;       // wave id 0..7 -> p sub-tile
  const int lane = tid & 31;    // wave32 lane
  const int nl = lane & 15;     // N / M low index
  const int hi = lane >> 4;     // upper-half-lane flag
  const int pw = 16 * w;        // p sub-tile base within 128-wide block tile

  // C (D-layout): VGPR r, lane n -> (g = r + 8*hi, p = n&15)
  v8f c;
#pragma unroll
  for (int r = 0; r < 8; ++r)
    c[r] = -2.0f * lds_gi[(pw + nl) * 16 + (r + 8 * hi)];

#pragma unroll
  for (int t = 0; t < 4; ++t) {
    const int k0 = 2 * hi;  // A/B VGPR0 holds K = 0 (low lanes) / 2 (high lanes)
    v2f a, bb;
    a[0] = lds_cd[(pw + 4 * t + k0 + 0) * 16 + nl];  // A[g=nl][k0]
    a[1] = lds_cd[(pw + 4 * t + k0 + 1) * 16 + nl];  // A[g=nl][k0+1]
    bb[0] = (nl == 4 * t + k0 + 0) ? 1.0f : 0.0f;    // B[k0][n=nl]
    bb[1] = (nl == 4 * t + k0 + 1) ? 1.0f : 0.0f;    // B[k0+1][n=nl]
    c = wmma_f32_16x16x4(a, bb, c);
  }

  // Store D: element (g = g0 + r + 8*hi, p = p0 + 16*w + nl)
#pragma unroll
  for (int r = 0; r < 8; ++r) {
    const size_t o =
        ((size_t)b * GN + (size_t)(g0 + r + 8 * hi)) * (size_t)PN +
        (size_t)(p0 + pw + nl);
    costT[o] = c[r];
  }
}

// ===========================================================================
// Kernel 2: exact Jonker-Volgenant LSA, one workgroup (32 waves) per batch.
// Mirrors the reference's augment_row loop op-for-op (same float association,
// same lowest-index argmin tie-break) so the discrete matching is identical.
// All mutable state lives in LDS (~49 KB << 320 KB/WGP).
// ===========================================================================
__device__ __forceinline__ unsigned long long shfl_xor_u64(unsigned long long x,
                                                           int m) {
  const int lo = __shfl_xor((int)(unsigned)(x & 0xffffffffull), m, 32);
  const int hi = __shfl_xor((int)(unsigned)(x >> 32), m, 32);
  return ((unsigned long long)(unsigned)hi << 32) | (unsigned)lo;
}

__device__ __forceinline__ unsigned long long pack_key(float f, int j) {
  unsigned u = __float_as_uint(f);
  u ^= (u & 0x80000000u) ? 0xffffffffu : 0x80000000u;  // total order on floats
  return ((unsigned long long)u << 32) | (unsigned)j;  // ties -> lowest index
}

__global__ __launch_bounds__(1024) void lsa_kernel(
    const float* __restrict__ costT, const int* __restrict__ nact_arr,
    float* __restrict__ out) {
  __shared__ float sh_shortest[PN];        // 16 KB
  __shared__ float sh_v[PN];               // 16 KB
  __shared__ short sh_row4col[PN];         // 8 KB
  __shared__ unsigned char sh_SC[PN];      // 4 KB
  __shared__ unsigned char sh_path[PN];    // 4 KB (row ids < 96 fit in u8)
  __shared__ float sh_u[GN];
  __shared__ short sh_col4row[GN];
  __shared__ unsigned char sh_SR[GN];
  __shared__ unsigned long long sh_wmin[32];
  __shared__ float s_minval;
  __shared__ int s_cur, s_sink;

  const int tid = threadIdx.x;
  const int b = blockIdx.x;
  const float* Cb = costT + (size_t)b * GN * PN;
  const int nact = nact_arr[b];

  // Zero this batch's output slices + init solver state.
  for (int j = tid; j < PN; j += 1024) {
    out[(size_t)b * PN + j] = 0.0f;                         // per_prop_gt_inds
    out[(size_t)BN * PN + (size_t)b * PN + j] = 0.0f;       // matched mask
    sh_v[j] = 0.0f;
    sh_row4col[j] = -1;
  }
  if (tid < GN) {
    sh_u[tid] = 0.0f;
    sh_col4row[tid] = -1;
  }
  __syncthreads();

  for (int i = 0; i < nact; ++i) {
    for (int j = tid; j < PN; j += 1024) {
      sh_shortest[j] = BIGF;
      sh_SC[j] = 0;
      sh_path[j] = 0;
    }
    if (tid < GN) sh_SR[tid] = 0;
    if (tid == 0) {
      s_minval = 0.0f;
      s_cur = i;
      s_sink = -1;
    }
    __syncthreads();

    // Shortest augmenting path.
    for (;;) {
      const int cur = s_cur;
      const float mv = s_minval;
      if (tid == 0) sh_SR[cur] = 1;
      const float ucur = sh_u[cur];
      const float* row = Cb + (size_t)cur * PN;  // coalesced, L2-resident

      unsigned long long lk = ~0ull;
#pragma unroll
      for (int k = 0; k < 4; ++k) {
        const int j = tid + k * 1024;
        if (!sh_SC[j]) {
          // same association as reference: ((minval + cost) - u[cur]) - v
          const float red = ((mv + row[j]) - ucur) - sh_v[j];
          if (red < sh_shortest[j]) {
            sh_shortest[j] = red;
            sh_path[j] = (unsigned char)cur;
          }
          const unsigned long long key = pack_key(sh_shortest[j], j);
          lk = key < lk ? key : lk;
        }
      }
      // wave32 min-reduce, then cross-wave via LDS.
#pragma unroll
      for (int m = 16; m > 0; m >>= 1) {
        const unsigned long long o = shfl_xor_u64(lk, m);
        lk = o < lk ? o : lk;
      }
      if ((tid & 31) == 0) sh_wmin[tid >> 5] = lk;
      __syncthreads();
      if (tid < 32) {
        unsigned long long z = sh_wmin[tid];
#pragma unroll
        for (int m = 16; m > 0; m >>= 1) {
          const unsigned long long o = shfl_xor_u64(z, m);
          z = o < z ? o : z;
        }
        if (tid == 0) {
          const int js = (int)(unsigned)(z & 0xffffffffu);
          s_minval = sh_shortest[js];
          sh_SC[js] = 1;
          const int r = sh_row4col[js];
          if (r < 0) s_sink = js; else s_cur = r;
        }
      }
      __syncthreads();
      if (s_sink >= 0) break;
    }

    // Dual updates (exactly the reference's order of operations).
    const float mv = s_minval;
    const int sink = s_sink;
    if (tid == 0) sh_u[i] = sh_u[i] + mv;
    if (tid < GN && tid != i && sh_SR[tid]) {
      int cj = sh_col4row[tid];
      if (cj < 0) cj = 0;  // clip, as in reference
      sh_u[tid] = (sh_u[tid] + mv) - sh_shortest[cj];
    }
#pragma unroll
    for (int k = 0; k < 4; ++k) {
      const int j = tid + k * 1024;
      if (sh_SC[j]) sh_v[j] = sh_v[j] - (mv - sh_shortest[j]);
    }
    __syncthreads();

    // Augment: flip alternating path (serial pointer chase, tiny).
    if (tid == 0) {
      int j = sink;
      for (;;) {
        const int r = sh_path[j];
        sh_row4col[j] = (short)r;
        const int jn = sh_col4row[r];
        sh_col4row[r] = (short)j;
        if (r == i) break;
        j = jn;
      }
    }
    __syncthreads();
  }

  // Scatter outputs: gt index (as float) and matched mask per proposal.
  if (tid < nact) {
    const int p = sh_col4row[tid];
    out[(size_t)b * PN + p] = (float)tid;
    out[(size_t)BN * PN + (size_t)b * PN + p] = 1.0f;
  }
}

// ===========================================================================
extern "C" void kernel_launch(void* const* d_in, const int* in_sizes, int n_in,
                              void* d_out, int out_size, void* d_ws,
                              size_t ws_size, hipStream_t stream) {
  (void)in_sizes; (void)n_in; (void)out_size; (void)ws_size;
  const float* center_dist = (const float*)d_in[0];  // [B,P,G] f32
  const float* gious = (const float*)d_in[1];        // [B,P,G] f32
  const int* nactual_gt = (const int*)d_in[2];       // [B] i32
  float* out = (float*)d_out;                        // [B*P] inds ++ [B*P] mask
  float* costT = (float*)d_ws;                       // [B,G,P] f32 = 12.6 MB

  dim3 g1(PN / 128, GN / 16, BN);
  build_cost_kernel<<<g1, dim3(256), 0, stream>>>(center_dist, gious, costT);
  lsa_kernel<<<dim3(BN), dim3(1024), 0, stream>>>(costT, nactual_gt, out);
}